// EdgeNet_36498632081416
// MI455X (gfx1250) — compile-verified
//
#include <hip/hip_runtime.h>
#include <hip/hip_bf16.h>
#include <math.h>

typedef float v2f __attribute__((ext_vector_type(2)));
typedef float v4f __attribute__((ext_vector_type(4)));
typedef float v8f __attribute__((ext_vector_type(8)));

#define N_NODES 8192
#define N_EDGES 32768
#define FEAT 11      // HID + 3
#define HID 8

// ---------------------------------------------------------------------------
// Kernel 1: stream a dense column-one-hot incidence matrix R[N_NODES][N_EDGES]
// and record, per column e, the row n where R[n][e] != 0.  Pure bandwidth:
// 1 GiB per matrix, 128-bit non-temporal loads, one scattered store per edge.
// ---------------------------------------------------------------------------
__global__ void onehot_scan_kernel(const float* __restrict__ R, int* __restrict__ idx) {
  const long long totalVec4 = ((long long)N_NODES * (long long)N_EDGES) >> 2;
  long long t = (long long)blockIdx.x * blockDim.x + threadIdx.x;
  const long long stride = (long long)gridDim.x * blockDim.x;
  const v4f* __restrict__ R4 = (const v4f*)R;
  for (long long i = t; i < totalVec4; i += stride) {
    v4f v = __builtin_nontemporal_load(R4 + i);
    if (v.x != 0.0f || v.y != 0.0f || v.z != 0.0f || v.w != 0.0f) {
      long long f = i << 2;                 // flat float index
      int n = (int)(f >> 15);               // row (node id): E = 2^15
      int e = (int)(f & (N_EDGES - 1));     // col (edge id)
      if (v.x != 0.0f) idx[e + 0] = n;
      if (v.y != 0.0f) idx[e + 1] = n;
      if (v.z != 0.0f) idx[e + 2] = n;
      if (v.w != 0.0f) idx[e + 3] = n;
    }
  }
}

// Branch-free concatenated edge feature:
//   K <  11 -> X[io][K]
//   K <  22 -> X[ii][K-11]
//   K >= 22 -> 0  (address clamped in-bounds, value selected to zero)
__device__ __forceinline__ float edge_feat(const float* __restrict__ X,
                                           int io, int ii, int K) {
  const bool second = (K >= FEAT);
  const bool pad    = (K >= 2 * FEAT);
  int kk = second ? (K - FEAT) : K;
  kk     = pad ? 0 : kk;                 // in-bounds always
  const int node = second ? ii : io;
  const float v  = X[node * FEAT + kk];  // unconditional load
  return pad ? 0.0f : v;                 // v_cndmask, no exec branching
}

// Branch-free padded W1 read: W1 is [22 x 8] row-major, padded to [24 x 16].
__device__ __forceinline__ float w1_val(const float* __restrict__ W1, int K, int col) {
  const bool pad = (K >= 2 * FEAT) | (col >= HID);
  const int kk = (K  < 2 * FEAT) ? K   : 0;
  const int cc = (col < HID)     ? col : 0;
  const float v = W1[kk * HID + cc];     // unconditional load
  return pad ? 0.0f : v;
}

// ---------------------------------------------------------------------------
// Kernel 2: per-edge MLP via V_WMMA_F32_16X16X4_F32.
// One wave32 handles a 16-edge tile:
//   A (16x4 f32)  : edge features, lanes 0-15 hold K={k0,k0+1}, lanes 16-31 K={k0+2,k0+3}
//   B (4x16 f32)  : W1 (8 real cols, padded to 16 with zeros), same K striping
//   C (16x16 f32) : VGPR r = row M (lanes 0-15: M=r, lanes 16-31: M=r+8), lane = col N
// Second layer done as a 16-lane butterfly dot with W2.
// ---------------------------------------------------------------------------
__global__ void edge_mlp_wmma_kernel(const float* __restrict__ X,
                                     const float* __restrict__ W1,
                                     const float* __restrict__ b1,
                                     const float* __restrict__ W2,
                                     const float* __restrict__ b2,
                                     const int* __restrict__ idx_o,
                                     const int* __restrict__ idx_i,
                                     float* __restrict__ out) {
  const int lane      = threadIdx.x & 31;
  const int waveInBlk = threadIdx.x >> 5;
  const int tile      = blockIdx.x * (blockDim.x >> 5) + waveInBlk;
  const int eBase     = tile * 16;
  const int nIdx      = lane & 15;  // A: row M this lane feeds; B/C: column N this lane owns
  const int hiHalf    = lane >> 4;  // selects K sub-pair within the 4-wide K step

  const int e  = eBase + nIdx;
  const int io = idx_o[e];
  const int ii = idx_i[e];

  // Fetch all A/B fragments up front (unconditional, clause-able loads),
  // then run the 6-deep WMMA accumulation back-to-back.
  float av[12], bv[12];
#pragma unroll
  for (int t = 0; t < 6; ++t) {
    const int ka = t * 4 + hiHalf * 2;
    av[2 * t + 0] = edge_feat(X, io, ii, ka + 0);
    av[2 * t + 1] = edge_feat(X, io, ii, ka + 1);
    bv[2 * t + 0] = w1_val(W1, ka + 0, nIdx);
    bv[2 * t + 1] = w1_val(W1, ka + 1, nIdx);
  }

  v8f acc = {0.f, 0.f, 0.f, 0.f, 0.f, 0.f, 0.f, 0.f};
#pragma unroll
  for (int t = 0; t < 6; ++t) {
    v2f a = {av[2 * t], av[2 * t + 1]};
    v2f b = {bv[2 * t], bv[2 * t + 1]};
    acc = __builtin_amdgcn_wmma_f32_16x16x4_f32(
        /*neg_a=*/false, a, /*neg_b=*/false, b,
        /*c_mod=*/(short)0, acc, /*reuse_a=*/false, /*reuse_b=*/false);
  }

  // Layer 1 activation + layer 2 dot (reduce over N = hidden dim; padded cols are 0).
  const int  cc    = (nIdx < HID) ? nIdx : 0;
  const float b1v  = b1[cc];
  const float w2v  = W2[cc];
  const float bias1 = (nIdx < HID) ? b1v : 0.0f;
  const float w2    = (nIdx < HID) ? w2v : 0.0f;

  float sums[8];
#pragma unroll
  for (int r = 0; r < 8; ++r) {
    float h = tanhf(acc[r] + bias1);
    float p = h * w2;
    p += __shfl_xor(p, 1, 32);
    p += __shfl_xor(p, 2, 32);
    p += __shfl_xor(p, 4, 32);
    p += __shfl_xor(p, 8, 32);   // lanes 0-15 all hold row-r sum; 16-31 hold row r+8
    sums[r] = p;
  }

  if (nIdx == 0) {               // lane 0 writes rows 0-7, lane 16 writes rows 8-15
    const int rowBase = eBase + hiHalf * 8;
    const float bb = b2[0];
#pragma unroll
    for (int r = 0; r < 8; ++r) {
      float s = sums[r] + bb;
      out[rowBase + r] = 1.0f / (1.0f + __expf(-s));
    }
  }
}

// ---------------------------------------------------------------------------
// setup_inputs order: X, Ri, Ro, W1, b1, W2, b2   (all float32)
// out: sigmoid scores, [N_EDGES] float32
// ---------------------------------------------------------------------------
extern "C" void kernel_launch(void* const* d_in, const int* in_sizes, int n_in,
                              void* d_out, int out_size, void* d_ws, size_t ws_size,
                              hipStream_t stream) {
  const float* X  = (const float*)d_in[0];
  const float* Ri = (const float*)d_in[1];
  const float* Ro = (const float*)d_in[2];
  const float* W1 = (const float*)d_in[3];
  const float* b1 = (const float*)d_in[4];
  const float* W2 = (const float*)d_in[5];
  const float* b2 = (const float*)d_in[6];
  float* out = (float*)d_out;

  int* idx_o = (int*)d_ws;            // [N_EDGES]
  int* idx_i = idx_o + N_EDGES;       // [N_EDGES]  (256 KB total scratch)

  // Bandwidth phase: 2 x 1 GiB streaming scans (each saturates HBM on its own).
  onehot_scan_kernel<<<8192, 256, 0, stream>>>(Ro, idx_o);  // bo <- Ro gather
  onehot_scan_kernel<<<8192, 256, 0, stream>>>(Ri, idx_i);  // bi <- Ri gather

  // Compute phase: 2048 16-edge WMMA tiles, 8 waves per 256-thread block.
  edge_mlp_wmma_kernel<<<(N_EDGES / 16) / 8, 256, 0, stream>>>(
      X, W1, b1, W2, b2, idx_o, idx_i, out);
}